// _PlainChebBranch_58488864637087
// MI455X (gfx1250) — compile-verified
//
#include <hip/hip_runtime.h>
#include <hip/hip_bf16.h>

typedef float v2f __attribute__((ext_vector_type(2)));
typedef float v8f __attribute__((ext_vector_type(8)));

#define DFEAT 128

// ---------------- fill ----------------

__global__ void fill_zero_i(int* __restrict__ p, size_t n) {
    size_t i = (size_t)blockIdx.x * blockDim.x + threadIdx.x;
    if (i < n) p[i] = 0;
}

// ---------------- degree counting ----------------
// deg_i[r]: all edges with source row r (CSR bucket sizes, self-loops included)
// deg_f[r]: masked degree (row != col) for the symmetric normalization

__global__ void count_deg(const int* __restrict__ row, const int* __restrict__ col,
                          int* __restrict__ deg_i, float* __restrict__ deg_f, int E) {
    int e = blockIdx.x * blockDim.x + threadIdx.x;
    if (e >= E) return;
    int r = row[e];
    atomicAdd(&deg_i[r], 1);
    if (r != col[e]) atomicAdd(&deg_f[r], 1.0f);
}

__global__ void dinv_kernel(const float* __restrict__ deg, float* __restrict__ dinv, int N) {
    int i = blockIdx.x * blockDim.x + threadIdx.x;
    if (i >= N) return;
    float d = deg[i];
    dinv[i] = (d > 0.f) ? rsqrtf(fmaxf(d, 1.0f)) : 0.f;
}

// ---------------- single-block exclusive scan (row_ptr) ----------------
// 1024 threads, Hillis-Steele in LDS, sequential chunks with running carry.

__global__ __launch_bounds__(1024) void scan_rowptr(const int* __restrict__ cnt,
                                                    int* __restrict__ rp, int N) {
    __shared__ int sh[1024];
    __shared__ int carry;
    if (threadIdx.x == 0) carry = 0;
    __syncthreads();
    for (int base = 0; base < N; base += 1024) {
        int i = base + (int)threadIdx.x;
        int v = (i < N) ? cnt[i] : 0;
        sh[threadIdx.x] = v;
        __syncthreads();
        for (int off = 1; off < 1024; off <<= 1) {
            int t = (threadIdx.x >= (unsigned)off) ? sh[threadIdx.x - off] : 0;
            __syncthreads();
            sh[threadIdx.x] += t;
            __syncthreads();
        }
        int incl  = sh[threadIdx.x];
        int total = sh[1023];
        if (i < N) rp[i] = carry + (incl - v);   // exclusive
        __syncthreads();
        if (threadIdx.x == 0) carry += total;
        __syncthreads();
    }
    if (threadIdx.x == 0) rp[N] = carry;         // == E
}

// ---------------- CSR build: one atomic per edge ----------------

__global__ void build_csr(const int* __restrict__ row, const int* __restrict__ col,
                          const float* __restrict__ dinv,
                          const int* __restrict__ rp, int* __restrict__ cursor,
                          int* __restrict__ csr_col, float* __restrict__ csr_w, int E) {
    int e = blockIdx.x * blockDim.x + threadIdx.x;
    if (e >= E) return;
    int r = row[e], c = col[e];
    int slot = rp[r] + atomicAdd(&cursor[r], 1);
    csr_col[slot] = c;
    csr_w[slot]   = (r != c) ? (-dinv[r] * dinv[c]) : 0.f;
}

// ---------------- SpMM gather: dst[r,:] = sum_e w[e] * src[col[e],:] ----------------
// One wave per destination row; 32 lanes x float4 = 128 features in registers.
// No atomics, no pre-zeroing; gathers are coalesced 512B/wave and L2-resident.

__global__ void spmm_gather(const float* __restrict__ src,
                            const int* __restrict__ rp,
                            const int* __restrict__ csr_col,
                            const float* __restrict__ csr_w,
                            float* __restrict__ dst, int N) {
    int gid  = blockIdx.x * blockDim.x + threadIdx.x;
    int r    = gid >> 5;
    int lane = threadIdx.x & 31;
    if (r >= N) return;
    int beg = rp[r], end = rp[r + 1];
    float4 acc = make_float4(0.f, 0.f, 0.f, 0.f);
    for (int i = beg; i < end; ++i) {
        int   c = csr_col[i];
        float w = csr_w[i];
        float4 v = *(const float4*)(src + (size_t)c * DFEAT + lane * 4);
        acc.x += w * v.x;
        acc.y += w * v.y;
        acc.z += w * v.z;
        acc.w += w * v.w;
    }
    *(float4*)(dst + (size_t)r * DFEAT + lane * 4) = acc;
}

// ---------------- fused Chebyshev GEMM epilogue ----------------
// Hout = relu(Hin@W[0] + T1@W[1] + (2*T2 - Hin)@W[2] + bias)
// One wave per 16-row block; 8 column tiles (16x16) in v8f acc[8]; fp32 WMMA 16x16x4.

template <int MODE> // 0: A = S0 ; 1: A = 2*S0 - S1
__device__ __forceinline__ void gemm_acc(const float* __restrict__ S0,
                                         const float* __restrict__ S1,
                                         const float* __restrict__ Wk,
                                         int row0, int m_lane, int half,
                                         v8f acc[8], int N) {
    int ar = row0 + m_lane;
    if (ar >= N) ar = N - 1; // clamp; OOB rows discarded at store
    const float* a0 = S0 + (size_t)ar * DFEAT;
    const float* a1 = (MODE == 1) ? (S1 + (size_t)ar * DFEAT) : nullptr;
    for (int kb = 0; kb < 32; ++kb) {
        int k0 = kb * 4 + half * 2;
        v2f a;
        if (MODE == 0) {
            a.x = a0[k0];
            a.y = a0[k0 + 1];
        } else {
            a.x = 2.0f * a0[k0]     - a1[k0];
            a.y = 2.0f * a0[k0 + 1] - a1[k0 + 1];
        }
#pragma unroll
        for (int ct = 0; ct < 8; ++ct) {
            v2f b;
            b.x = Wk[(size_t)k0 * DFEAT + ct * 16 + m_lane];
            b.y = Wk[(size_t)(k0 + 1) * DFEAT + ct * 16 + m_lane];
            acc[ct] = __builtin_amdgcn_wmma_f32_16x16x4_f32(
                false, a, false, b, (short)0, acc[ct], false, false);
        }
    }
}

__global__ __launch_bounds__(256) void cheb_wmma_fused(
    const float* __restrict__ Hin, const float* __restrict__ T1,
    const float* __restrict__ T2, const float* __restrict__ W, // [3,128,128]
    const float* __restrict__ bias, float* __restrict__ Hout, int N) {

    int wave = threadIdx.x >> 5;        // wave-uniform
    int rb   = blockIdx.x * 8 + wave;   // wave-uniform row block
    int nRB  = (N + 15) >> 4;
    if (rb >= nRB) return;              // uniform branch: EXEC stays all-1s for WMMA

    int lane   = threadIdx.x & 31;
    int m_lane = lane & 15;
    int half   = lane >> 4;
    int row0   = rb << 4;

    v8f acc[8];
#pragma unroll
    for (int ct = 0; ct < 8; ++ct) acc[ct] = (v8f){0.f,0.f,0.f,0.f,0.f,0.f,0.f,0.f};

    gemm_acc<0>(Hin, nullptr, W,                     row0, m_lane, half, acc, N); // Tx0 @ W0
    gemm_acc<0>(T1,  nullptr, W + DFEAT * DFEAT,     row0, m_lane, half, acc, N); // Tx1 @ W1
    gemm_acc<1>(T2,  Hin,     W + 2 * DFEAT * DFEAT, row0, m_lane, half, acc, N); // (2*T2-Tx0)@W2

    // epilogue: + bias, relu, store. C/D layout: n = lane%16, m = v + 8*half
#pragma unroll
    for (int ct = 0; ct < 8; ++ct) {
        int n = ct * 16 + m_lane;
        float bv = bias[n];
#pragma unroll
        for (int v = 0; v < 8; ++v) {
            int r = row0 + v + 8 * half;
            if (r < N) {
                float val = acc[ct][v] + bv;
                Hout[(size_t)r * DFEAT + n] = fmaxf(val, 0.f);
            }
        }
    }
}

// ---------------- host orchestration ----------------

extern "C" void kernel_launch(void* const* d_in, const int* in_sizes, int n_in,
                              void* d_out, int out_size, void* d_ws, size_t ws_size,
                              hipStream_t stream) {
    const float* x  = (const float*)d_in[0];
    const int*   ei = (const int*)d_in[1];
    const float* W1 = (const float*)d_in[2];
    const float* b1 = (const float*)d_in[3];
    const float* W2 = (const float*)d_in[4];
    const float* b2 = (const float*)d_in[5];
    const float* W3 = (const float*)d_in[6];
    const float* b3 = (const float*)d_in[7];

    const int N = in_sizes[0] / DFEAT;
    const int E = in_sizes[1] / 2;
    const int* row = ei;
    const int* col = ei + E;

    char* ws = (char*)d_ws;
    size_t off = 0;
    auto alloc = [&](size_t elems) { void* p = ws + off; off += elems * 4; return p; };

    // zero-init region: deg_f | deg_i | cursor (3N dwords, contiguous)
    float* deg_f  = (float*)alloc(N);
    int*   deg_i  = (int*)  alloc(N);
    int*   cursor = (int*)  alloc(N);
    float* dinv   = (float*)alloc(N);
    int*   rp     = (int*)  alloc((size_t)N + 1);
    int*   csr_c  = (int*)  alloc(E);
    float* csr_w  = (float*)alloc(E);
    float* t1     = (float*)alloc((size_t)N * DFEAT);
    float* t2     = (float*)alloc((size_t)N * DFEAT);
    float* hA     = (float*)alloc((size_t)N * DFEAT);
    float* hB     = (float*)alloc((size_t)N * DFEAT);

    // ---- CSR + Chebyshev edge weights (once per call) ----
    {
        size_t zn = (size_t)3 * N; // deg_f, deg_i, cursor are contiguous
        fill_zero_i<<<(unsigned)((zn + 255) / 256), 256, 0, stream>>>((int*)deg_f, zn);
        count_deg<<<(E + 255) / 256, 256, 0, stream>>>(row, col, deg_i, deg_f, E);
        dinv_kernel<<<(N + 255) / 256, 256, 0, stream>>>(deg_f, dinv, N);
        scan_rowptr<<<1, 1024, 0, stream>>>(deg_i, rp, N);
        build_csr<<<(E + 255) / 256, 256, 0, stream>>>(row, col, dinv, rp, cursor,
                                                       csr_c, csr_w, E);
    }

    const int nRB      = (N + 15) / 16;
    const int gemmBlks = (nRB + 7) / 8;
    const int rowBlks  = (N + 7) / 8;   // 8 row-waves per 256-thread block

    auto layer = [&](const float* Hin, const float* W, const float* bias, float* Hout) {
        spmm_gather<<<rowBlks, 256, 0, stream>>>(Hin, rp, csr_c, csr_w, t1, N);
        spmm_gather<<<rowBlks, 256, 0, stream>>>(t1,  rp, csr_c, csr_w, t2, N);
        cheb_wmma_fused<<<gemmBlks, 256, 0, stream>>>(Hin, t1, t2, W, bias, Hout, N);
    };

    layer(x,  W1, b1, hA);
    layer(hA, W2, b2, hB);
    layer(hB, W3, b3, (float*)d_out);
}